// LocalAttentionLayer_16114717295301
// MI455X (gfx1250) — compile-verified
//
#include <hip/hip_runtime.h>

typedef float v2f __attribute__((ext_vector_type(2)));
typedef float v8f __attribute__((ext_vector_type(8)));
typedef int   v4i __attribute__((vector_size(16)));   // pointee type expected by async-LDS builtin

#if defined(__HIP_DEVICE_COMPILE__) && __has_builtin(__builtin_amdgcn_global_load_async_to_lds_b128)
#define USE_ASYNC 1
#else
#define USE_ASYNC 0
#endif

namespace {
constexpr int L  = 1024;
constexpr int M  = 16;
constexpr int S  = 64;
constexpr int E  = 128;
constexpr int EV = 128;
constexpr int WAVES = 2;           // waves per block, one (l,m) per wave
constexpr int KSTRIDE = 132;       // padded LDS row stride (floats): 4-bank rotation per row
constexpr int TILE_FLOATS = 16 * KSTRIDE;
constexpr float MASK_VALUE = -10000000.0f;
}

// D = A(16x4 f32) * B(4x16 f32) + C(16x16 f32), wave32 WMMA.
__device__ __forceinline__ v8f wmma_f32_16x16x4(v2f a, v2f b, v8f c) {
#if defined(__HIP_DEVICE_COMPILE__)
    return __builtin_amdgcn_wmma_f32_16x16x4_f32(false, a, false, b,
                                                 (short)0, c, false, false);
#else
    (void)a; (void)b;
    return c;
#endif
}

// Stage one 16x128 f32 tile (8 KB) into LDS with padded row stride.
// Async path: 16x GLOBAL_LOAD_ASYNC_TO_LDS_B128 (512B per issue across the wave),
// no VGPR round-trip, tracked by ASYNCcnt. Fallback: b128 load + ds_store.
__device__ __forceinline__ void stage_tile(const float* __restrict__ gsrc,
                                           float* __restrict__ lds_tile, int lane) {
#if USE_ASYNC
    float* g = const_cast<float*>(gsrc);
    #pragma unroll
    for (int i = 0; i < 16; ++i) {
        __builtin_amdgcn_global_load_async_to_lds_b128(
            (__attribute__((address_space(1))) v4i*)(g + i * 128 + lane * 4),
            (__attribute__((address_space(3))) v4i*)(lds_tile + i * KSTRIDE + lane * 4),
            0, 0);
    }
#else
    __builtin_prefetch((const char*)gsrc + 8192 + lane * 256, 0, 0);
    const float4* g4 = (const float4*)gsrc;
    #pragma unroll
    for (int i = 0; i < 16; ++i) {
        float4 d = g4[i * 32 + lane];
        *(float4*)&lds_tile[i * KSTRIDE + lane * 4] = d;
    }
#endif
}

// Wait until at most N async copies are outstanding (async loads complete in
// order, so N==16 means "previous tile fully in LDS while next still flies").
template <int N>
__device__ __forceinline__ void wait_staged() {
#if USE_ASYNC
#if __has_builtin(__builtin_amdgcn_s_wait_asynccnt)
    __builtin_amdgcn_s_wait_asynccnt(N);
#else
    asm volatile("s_wait_asynccnt %0" :: "i"(N) : "memory");
#endif
#endif
    __syncthreads();   // fallback visibility; uniform across both waves
}

__global__ __launch_bounds__(WAVES * 32)
void local_attn_wmma_kernel(const float* __restrict__ q,
                            const float* __restrict__ keys,
                            const float* __restrict__ values,
                            const unsigned char* __restrict__ mask,
                            float* __restrict__ out_pooled,
                            float* __restrict__ out_w)
{
    __shared__ __align__(16) float q_sm[E];
    __shared__ __align__(16) float tile_sm[WAVES][2][TILE_FLOATS]; // double-buffered K/V tiles
    __shared__ float w_sm[WAVES][S];                               // scores -> softmax weights

    const int tid  = threadIdx.x;
    const int lane = tid & 31;
    const int wave = tid >> 5;
    const int l  = blockIdx.x;
    const int m  = blockIdx.y * WAVES + wave;
    const int lm = l * M + m;

    float* buf0 = tile_sm[wave][0];
    float* buf1 = tile_sm[wave][1];
    float* wsm  = w_sm[wave];

    // ISA A/B fragment layout for 16x4 f32: lanes 0-15 carry K=k0,k0+1 ; lanes 16-31 carry K=k0+2,k0+3
    const int half = (lane < 16) ? 0 : 2;
    const int nlo  = lane & 15;

    for (int t = tid; t < E; t += WAVES * 32) q_sm[t] = q[(size_t)l * E + t];
    __syncthreads();

    const float* kbase = keys   + (size_t)lm * S * E;
    const float* vbase = values + (size_t)lm * S * EV;

    // ---------------- Phase 1: scores = K_tile @ q (q replicated across N columns) ----------------
    stage_tile(kbase, buf0, lane);
    for (int st = 0; st < 4; ++st) {
        float* cur = (st & 1) ? buf1 : buf0;
        if (st < 3) {
            stage_tile(kbase + (size_t)(st + 1) * 16 * E, (st & 1) ? buf0 : buf1, lane);
            wait_staged<16>();
        } else {
            wait_staged<0>();
        }

        v8f acc = {0.f, 0.f, 0.f, 0.f, 0.f, 0.f, 0.f, 0.f};
        #pragma unroll 8
        for (int kc = 0; kc < 32; ++kc) {
            const int k0 = kc * 4 + half;
            v2f a, b;
            a.x = cur[nlo * KSTRIDE + k0];
            a.y = cur[nlo * KSTRIDE + k0 + 1];
            b.x = q_sm[k0];
            b.y = q_sm[k0 + 1];
            acc = wmma_f32_16x16x4(a, b, acc);
        }
        // D: VGPR r = row r (lanes 0-15) / row r+8 (lanes 16-31); all N columns identical.
        if (nlo == 0) {
            const int rbase = st * 16 + ((lane >= 16) ? 8 : 0);
            #pragma unroll
            for (int r = 0; r < 8; ++r) wsm[rbase + r] = acc[r];
        }
    }

    // Kick off the first V tile so its latency hides under the softmax.
    stage_tile(vbase, buf0, lane);
    __syncthreads();

    // ---------------- Masked softmax over S=64 (2 elements / lane) ----------------
    const unsigned char* mrow = mask + (size_t)lm * S;
    float x0 = wsm[lane]      + (mrow[lane]      ? 0.f : MASK_VALUE);
    float x1 = wsm[lane + 32] + (mrow[lane + 32] ? 0.f : MASK_VALUE);
    float mx = fmaxf(x0, x1);
    #pragma unroll
    for (int off = 16; off > 0; off >>= 1) mx = fmaxf(mx, __shfl_xor(mx, off, 32));
    float e0 = __expf(x0 - mx);
    float e1 = __expf(x1 - mx);
    float sum = e0 + e1;
    #pragma unroll
    for (int off = 16; off > 0; off >>= 1) sum += __shfl_xor(sum, off, 32);
    const float inv = 1.0f / sum;
    const float w0 = e0 * inv, w1 = e1 * inv;
    wsm[lane]      = w0;
    wsm[lane + 32] = w1;
    out_w[(size_t)lm * S + lane]      = w0;
    out_w[(size_t)lm * S + lane + 32] = w1;

    // ---------------- Phase 2: pooled = w @ V (w replicated across all 16 M rows) ----------------
    v8f acc2[8];
    #pragma unroll
    for (int nc = 0; nc < 8; ++nc)
        acc2[nc] = (v8f){0.f, 0.f, 0.f, 0.f, 0.f, 0.f, 0.f, 0.f};

    for (int st = 0; st < 4; ++st) {
        float* cur = (st & 1) ? buf1 : buf0;
        if (st < 3) {
            stage_tile(vbase + (size_t)(st + 1) * 16 * EV, (st & 1) ? buf0 : buf1, lane);
            wait_staged<16>();
        } else {
            wait_staged<0>();
        }

        #pragma unroll
        for (int kc = 0; kc < 4; ++kc) {
            const int kl = kc * 4 + half;           // row within tile for this lane's K slots
            v2f a;
            a.x = wsm[st * 16 + kl];
            a.y = wsm[st * 16 + kl + 1];
            #pragma unroll
            for (int nc = 0; nc < 8; ++nc) {
                const int n = nc * 16 + nlo;
                v2f b;
                b.x = cur[kl * KSTRIDE + n];
                b.y = cur[(kl + 1) * KSTRIDE + n];
                acc2[nc] = wmma_f32_16x16x4(a, b, acc2[nc]);
            }
        }
    }

    // All rows of D identical; VGPR0 lane n (n = lane&15) holds pooled[nc*16 + n].
    if (lane < 16) {
        #pragma unroll
        for (int nc = 0; nc < 8; ++nc)
            out_pooled[(size_t)lm * EV + nc * 16 + lane] = acc2[nc][0];
    }
}

extern "C" void kernel_launch(void* const* d_in, const int* in_sizes, int n_in,
                              void* d_out, int out_size, void* d_ws, size_t ws_size,
                              hipStream_t stream) {
    (void)in_sizes; (void)n_in; (void)out_size; (void)d_ws; (void)ws_size;
    const float* q            = (const float*)d_in[0];
    const float* keys         = (const float*)d_in[1];
    const float* values       = (const float*)d_in[2];
    const unsigned char* mask = (const unsigned char*)d_in[3];  // jnp.bool_ -> 1 byte/elem

    float* out_pooled = (float*)d_out;                          // (L, M, EV) first
    float* out_w      = out_pooled + (size_t)L * M * EV;        // then (L, M, S)

    dim3 grid(L, M / WAVES, 1);
    dim3 block(WAVES * 32, 1, 1);
    hipLaunchKernelGGL(local_attn_wmma_kernel, grid, block, 0, stream,
                       q, keys, values, mask, out_pooled, out_w);
}